// MPEncoder_44719199485974
// MI455X (gfx1250) — compile-verified
//
#include <hip/hip_runtime.h>

// ---------------------------------------------------------------------------
// MPEncoder: 2-layer message-passing GNN on MI455X (gfx1250, wave32).
//   layer(h): s = segment_sum(h[src], dst); a = h + s/max(deg,1);
//             out = relu(a @ W + b)
// Roofline: edge scatter moves ~2.5 GB (dominant, ~110us @ 23.3 TB/s);
// GEMMs are only 6.5 GFLOP -> fp32 WMMA (16x16x4) keeps exact precision at
// zero cost. A-tiles (16 x K, shared by all 8 waves of a block) are staged
// into LDS by the Tensor Data Mover (TENSOR_LOAD_TO_LDS) with row padding
// for bank-conflict-free reads; tiles are read back with inline-asm
// ds_load_b64 (the compiler cannot see TDM's LDS writes). LDS is allocated
// as dynamic shared memory at launch so the group segment is really sized
// (a static array was dead-stripped since only opaque ops touch it).
// ---------------------------------------------------------------------------

typedef float v2f  __attribute__((ext_vector_type(2)));
typedef float v8f  __attribute__((ext_vector_type(8)));
typedef unsigned int u32;
typedef unsigned int u32x4 __attribute__((ext_vector_type(4)));
typedef int          i32x4 __attribute__((ext_vector_type(4)));
typedef int          i32x8 __attribute__((ext_vector_type(8)));

#define N_NODES 50000
#define DIM_IN  128
#define DIM_HID 256
#define DIM_OUT 128
#define LDS_PAD 4   // DWORDs of padding per row (TDM pad_amount code 3)

// ---------------------------------------------------------------- utilities
__global__ void zero_f32_kernel(float* __restrict__ p, long n) {
  long i = (long)blockIdx.x * blockDim.x + threadIdx.x;
  long stride = (long)gridDim.x * blockDim.x;
  for (; i < n; i += stride) p[i] = 0.0f;
}

__global__ void degree_kernel(const int* __restrict__ dst,
                              float* __restrict__ deg, int E) {
  int i = blockIdx.x * blockDim.x + threadIdx.x;
  if (i < E) atomicAdd(&deg[dst[i]], 1.0f);
}

__global__ void invdeg_kernel(float* __restrict__ deg, int n) {
  int i = blockIdx.x * blockDim.x + threadIdx.x;
  if (i < n) deg[i] = 1.0f / fmaxf(deg[i], 1.0f);
}

// ------------------------------------------------------- edge scatter-add
// One edge handled by D/4 lanes (float4 per lane). shift = log2(D/4).
__global__ void scatter_add_kernel(const float* __restrict__ feat,
                                   const int* __restrict__ src,
                                   const int* __restrict__ dst,
                                   float* __restrict__ acc,
                                   int E, int D, int shift) {
  long t = (long)blockIdx.x * blockDim.x + threadIdx.x;
  int e = (int)(t >> shift);
  if (e >= E) return;
  int c = ((int)t & ((1 << shift) - 1)) << 2;
  int sr = src[e];
  int ds = dst[e];
  const float4 v = *(const float4*)(feat + (long)sr * D + c);
  float* o = acc + (long)ds * D + c;
  atomicAdd(o + 0, v.x);
  atomicAdd(o + 1, v.y);
  atomicAdd(o + 2, v.z);
  atomicAdd(o + 3, v.w);
}

// --------------------------------------------- Tensor Data Mover descriptor
// 2-D fp32 tile load: 16 rows x K dwords, row-major tensor, LDS row padding
// of 4 DWORDs after every K DWORDs (pad_interval code: 6 -> 128, 7 -> 256).
// D# layout per CDNA5 ISA ch.8: group0 (4 dw), group1 (8 dw), groups 2/3
// zero (2-D tensor, VADDR2/3 conceptually NULL).
// This toolchain's builtin takes 6 args: (g0, g1, g2, g3, g_extra, cpol).
__device__ inline void tdm_load_tile_2d(u32 lds_byte_off, const void* gptr,
                                        u32 K, u32 pad_interval_code) {
  unsigned long long ga = (unsigned long long)(size_t)gptr;
  u32x4 g0;
  g0.x = 1u;                                   // count=1, user descriptor
  g0.y = lds_byte_off;                         // lds_addr (bytes)
  g0.z = (u32)(ga & 0xffffffffu);              // global_addr[31:0]
  g0.w = (u32)((ga >> 32) & 0x01ffffffu)       // global_addr[56:32]
       | (2u << 30);                           // type = 2 ("image")
  const u32 rows = 16u;
  u32 d0 = (2u << 16)                          // data_size = 4 bytes
         | (1u << 20)                          // pad_enable
         | (pad_interval_code << 22)           // DWORDs before pad
         | (3u << 25);                         // pad_amount = 4 DWORDs
  u32 d1 = (K & 0xffffu) << 16;                // tensor_dim0[15:0]
  u32 d2 = ((K >> 16) & 0xffffu)               // tensor_dim0[31:16]
         | ((rows & 0xffffu) << 16);           // tensor_dim1[15:0]
  u32 d3 = ((rows >> 16) & 0xffffu)            // tensor_dim1[31:16]
         | ((K & 0xffffu) << 16);              // tile_dim0 = K
  u32 d4 = rows & 0xffffu;                     // tile_dim1 = 16, tile_dim2 = 0
  u32 d5 = K;                                  // tensor_dim0_stride[31:0]
  i32x8 g1 = {(int)d0, (int)d1, (int)d2, (int)d3,
              (int)d4, (int)d5, 0, 0};         // dim1_stride = 0 (unused, 2D)
  i32x4 z4 = {0, 0, 0, 0};
  i32x8 z8 = {0, 0, 0, 0, 0, 0, 0, 0};
  __builtin_amdgcn_tensor_load_to_lds(g0, g1, z4, z4, z8, /*cpol=*/0);
}

// ------------------------------------------------------- fused WMMA GEMM
// OUT[nRows,Nout] = relu( (X + S * invdeg[row]) @ W + bias )
// Block = 8 waves; all 8 waves share one 16-row A stripe (TDM-staged in LDS)
// and compute 8 adjacent 16x16 N-tiles. K-loop of fp32 16x16x4 WMMAs.
// A frag (f32 16x4): lane L -> M = L&15, K pair = (L>>4)*2 .. +1 (2 VGPRs)
// B frag (f32 4x16): lane L -> N = L&15, same K pair (2 VGPRs)
// C/D (f32 16x16):   VGPR v -> M = v + 8*(L>>4), N = L&15 (8 VGPRs)
// LDS layout (dynamic shared, base offset 0): X tile | S tile, each
// 16 x (K + LDS_PAD) dwords.
__global__ __launch_bounds__(256) void gnn_gemm_kernel(
    const float* __restrict__ X, const float* __restrict__ S,
    const float* __restrict__ invdeg, const float* __restrict__ W,
    const float* __restrict__ bias, float* __restrict__ OUT,
    int K, int Nout, int tnGroups, u32 padIntervalCode) {
  const int lane = threadIdx.x & 31;
  const int wave = threadIdx.x >> 5;
  const int tm   = blockIdx.x / tnGroups;       // 16-row stripe index
  const int tng  = blockIdx.x % tnGroups;
  const int row0 = tm << 4;
  const int col0 = ((tng << 3) + wave) << 4;    // this wave's 16-col tile
  const int lstride = K + LDS_PAD;              // padded LDS row stride (dw)
  const u32 stileOff = (u32)(16 * lstride * 4); // S tile LDS byte offset

  if (wave == 0) {
    // TDM: stage 16xK X tile and S tile into LDS (shared by all 8 waves)
    tdm_load_tile_2d(0u, X + (long)row0 * K, (u32)K, padIntervalCode);
    tdm_load_tile_2d(stileOff, S + (long)row0 * K, (u32)K, padIntervalCode);
    __builtin_amdgcn_s_wait_tensorcnt(0);
  }
  __syncthreads();

  const int m  = lane & 15;         // A row within tile (both lane halves)
  const int kk = (lane >> 4) << 1;  // K offset within 4-step: 0 or 2
  const int n  = lane & 15;         // B/C column within tile
  const float idg = invdeg[row0 + m];
  // LDS byte base of this lane's row; banks 4*m mod 64 across lanes: clean
  const u32 xbase = (u32)((m * lstride + kk) * 4);
  const u32 sbase = xbase + stileOff;

  v8f acc = {0.f, 0.f, 0.f, 0.f, 0.f, 0.f, 0.f, 0.f};

  for (int kb = 0; kb < K; kb += 4) {
    const u32 xo = xbase + (u32)(kb * 4);
    const u32 so = sbase + (u32)(kb * 4);
    v2f xv, sv;
    // Opaque LDS reads: the compiler cannot see the TDM's LDS writes, so
    // plain loads would be folded away. Issue both, then one dscnt wait.
    asm volatile("ds_load_b64 %0, %2\n\t"
                 "ds_load_b64 %1, %3\n\t"
                 "s_wait_dscnt 0x0"
                 : "=&v"(xv), "=&v"(sv)
                 : "v"(xo), "v"(so));
    v2f a, b;
    a.x = fmaf(sv.x, idg, xv.x);  // fused A = X + S * (1/deg)
    a.y = fmaf(sv.y, idg, xv.y);
    // W row-major [K, Nout]; lanes 0-15 coalesce one row, 16-31 the next pair
    b.x = W[(long)(kb + kk) * Nout + col0 + n];
    b.y = W[(long)(kb + kk + 1) * Nout + col0 + n];
    acc = __builtin_amdgcn_wmma_f32_16x16x4_f32(
        /*neg_a=*/false, a, /*neg_b=*/false, b,
        /*c_mod=*/(short)0, acc, /*reuse_a=*/false, /*reuse_b=*/false);
  }

  const int cRowBase = row0 + ((lane >> 4) << 3);  // +0 or +8
  const int col = col0 + n;
  const float bv = bias[col];
#pragma unroll
  for (int v = 0; v < 8; ++v) {
    float r = acc[v] + bv;
    r = r > 0.0f ? r : 0.0f;  // relu
    OUT[(long)(cRowBase + v) * Nout + col] = r;
  }
}

// --------------------------------------------------------------- launcher
extern "C" void kernel_launch(void* const* d_in, const int* in_sizes, int n_in,
                              void* d_out, int out_size, void* d_ws, size_t ws_size,
                              hipStream_t stream) {
  const float* x   = (const float*)d_in[0];  // [N,128]
  const float* W1  = (const float*)d_in[1];  // [128,256]
  const float* b1  = (const float*)d_in[2];  // [256]
  const float* W2  = (const float*)d_in[3];  // [256,128]
  const float* b2  = (const float*)d_in[4];  // [128]
  const int* esrc  = (const int*)d_in[5];    // [E]
  const int* edst  = (const int*)d_in[6];    // [E]
  float* out = (float*)d_out;                // [N,128]
  const int E = in_sizes[5];
  const int N = N_NODES;

  // workspace layout: deg/invdeg [N] | h [N,256] | s [N,256]
  char* ws = (char*)d_ws;
  size_t off = 0;
  float* deg = (float*)(ws + off);
  off += ((size_t)N * 4 + 255) & ~(size_t)255;
  float* h = (float*)(ws + off);
  off += (size_t)N * DIM_HID * 4;
  float* s = (float*)(ws + off);

  const int ZB = 2048, ZT = 256;

  // --- degree + reciprocal -------------------------------------------------
  zero_f32_kernel<<<ZB, ZT, 0, stream>>>(deg, (long)N);
  degree_kernel<<<(E + 255) / 256, 256, 0, stream>>>(edst, deg, E);
  invdeg_kernel<<<(N + 255) / 256, 256, 0, stream>>>(deg, N);

  // --- layer 1: scatter x -> s[N,128], gemm -> h[N,256] --------------------
  zero_f32_kernel<<<ZB, ZT, 0, stream>>>(s, (long)N * DIM_IN);
  {
    long tot = (long)E * (DIM_IN / 4);  // 32 lanes/edge
    scatter_add_kernel<<<(int)((tot + 255) / 256), 256, 0, stream>>>(
        x, esrc, edst, s, E, DIM_IN, 5);
  }
  {
    // 3125 row stripes x 2 groups of 8 N-tiles (Nout=256 -> tilesN=16)
    int tnGroups = (DIM_HID / 16) / 8;                      // 2
    int blocks = (N / 16) * tnGroups;                       // 6250
    size_t ldsBytes = 2u * 16u * (DIM_IN + LDS_PAD) * 4u;   // 16896 B
    gnn_gemm_kernel<<<blocks, 256, ldsBytes, stream>>>(
        x, s, deg, W1, b1, h, DIM_IN, DIM_HID, tnGroups,
        /*pad_interval: 128 DWORDs=*/6u);
  }

  // --- layer 2: scatter h -> s[N,256], gemm -> out[N,128] ------------------
  zero_f32_kernel<<<ZB, ZT, 0, stream>>>(s, (long)N * DIM_HID);
  {
    long tot = (long)E * (DIM_HID / 4);  // 64 lanes/edge
    scatter_add_kernel<<<(int)((tot + 255) / 256), 256, 0, stream>>>(
        h, esrc, edst, s, E, DIM_HID, 6);
  }
  {
    int tnGroups = (DIM_OUT / 16) / 8;                      // 1
    int blocks = (N / 16) * tnGroups;                       // 3125
    size_t ldsBytes = 2u * 16u * (DIM_HID + LDS_PAD) * 4u;  // 33280 B
    gnn_gemm_kernel<<<blocks, 256, ldsBytes, stream>>>(
        h, s, deg, W2, b2, out, DIM_HID, DIM_OUT, tnGroups,
        /*pad_interval: 256 DWORDs=*/7u);
  }
}